// HeteroGatedGCNLayer_83296595739224
// MI455X (gfx1250) — compile-verified
//
#include <hip/hip_runtime.h>
#include <hip/hip_bf16.h>
#include <math.h>

#define D      256
#define STEPS  3
#define EPSLN  1e-5f

typedef float v2f __attribute__((ext_vector_type(2)));
typedef float v4f __attribute__((ext_vector_type(4)));
typedef float v8f __attribute__((ext_vector_type(8)));

// Toggle: 1 = CDNA5 async global->LDS copies (ASYNCcnt), 0 = sync VGPR staging
#ifndef USE_ASYNC_LDS
#define USE_ASYNC_LDS 1
#endif

__device__ __forceinline__ void cp16_g2l(float* l, const float* g)
{
#if USE_ASYNC_LDS
    // per-lane 16B copy, tracked by ASYNCcnt, no VGPR data round-trip
    unsigned laddr = (unsigned)(uintptr_t)l;   // low 32 bits = LDS byte offset
    asm volatile("global_load_async_to_lds_b128 %0, %1, off"
                 :: "v"(laddr), "v"(g)
                 : "memory");
#else
    *(v4f*)l = *(const v4f*)g;
#endif
}

__device__ __forceinline__ void async_wait_all()
{
#if USE_ASYNC_LDS
    asm volatile("s_wait_asynccnt 0x0" ::: "memory");
#endif
}

// ---------------------------------------------------------------------------
// GEMM:  C[M, Nout] = A[M, 256] * W[Nout, 256]^T + bias[Nout]      (fp32 WMMA)
// Block: 256 threads = 8 waves (2x4).  Block tile 128(M) x 128(N).
// Wave tile 64x32 = 4x2 V_WMMA_F32_16X16X4_F32 accumulators.
// Double-buffered LDS, async global->LDS staging, one barrier per K-chunk.
// ---------------------------------------------------------------------------
#define BM     128
#define BN     128
#define KC     16
#define NCHUNK (D / KC)             // 16
#define LDSROW 20                   // padded row stride (floats): 80B rows ->
                                    // 16B-aligned f4 stores, conflict-free f2 loads
#define ABUF   (BM * LDSROW)        // 2560 floats
#define BBUF   (BN * LDSROW)        // 2560 floats
#define BUFSZ  (ABUF + BBUF)        // 5120 floats = 20480 B per buffer

__global__ __launch_bounds__(256)
void gemm_bias_wmma(const float* __restrict__ A, const float* __restrict__ W,
                    const float* __restrict__ bias, float* __restrict__ C,
                    int M, int Nout)
{
    __shared__ float smem[2 * BUFSZ];      // 40960 B

    const int tid  = threadIdx.x;
    const int lane = tid & 31;
    const int wid  = tid >> 5;     // 0..7
    const int wm   = wid >> 2;     // 0..1 (M direction, 64 rows each)
    const int wn   = wid & 3;      // 0..3 (N direction, 32 cols each)
    const int r    = lane & 15;
    const int hi   = lane >> 4;

    const int nBase = blockIdx.x * BN;
    const int mBase = blockIdx.y * BM;

    v8f zero = {};
    v8f acc[4][2];
    #pragma unroll
    for (int i = 0; i < 4; i++)
        #pragma unroll
        for (int j = 0; j < 2; j++) acc[i][j] = zero;

    // stage one K-chunk (A: 128x16, B: 128x16) into LDS buffer `buf`
    auto stage = [&](int k0, float* buf) {
        float* bA = buf;
        float* bB = buf + ABUF;
        #pragma unroll
        for (int t = 0; t < 2; t++) {                 // A: 512 f4, 2/thread
            int idx = tid + t * 256;
            int row = idx >> 2, cc = (idx & 3) * 4;
            int grow = mBase + row; if (grow >= M) grow = M - 1;    // clamp tail
            cp16_g2l(bA + row * LDSROW + cc, A + (size_t)grow * D + k0 + cc);
        }
        #pragma unroll
        for (int t = 0; t < 2; t++) {                 // B: 512 f4, 2/thread
            int idx = tid + t * 256;
            int row = idx >> 2, cc = (idx & 3) * 4;
            cp16_g2l(bB + row * LDSROW + cc, W + (size_t)(nBase + row) * D + k0 + cc);
        }
    };

    // prologue: chunk 0 -> buffer 0
    stage(0, smem);
    async_wait_all();
    __syncthreads();

    for (int c = 0; c < NCHUNK; ++c) {
        const float* sA = smem + (c & 1) * BUFSZ;
        const float* sB = sA + ABUF;

        if (c + 1 < NCHUNK)                        // prefetch next chunk
            stage((c + 1) * KC, smem + ((c + 1) & 1) * BUFSZ);

        #pragma unroll
        for (int kk = 0; kk < KC; kk += 4) {
            v2f a[4], b[2];
            // A frag (16x4): lane r holds {A[r][k+2hi], A[r][k+2hi+1]}
            #pragma unroll
            for (int i = 0; i < 4; i++)
                a[i] = *(const v2f*)(sA + (wm * 64 + i * 16 + r) * LDSROW + kk + 2 * hi);
            // B frag (4x16): lane r holds column n=r, K pair selected by hi
            #pragma unroll
            for (int j = 0; j < 2; j++)
                b[j] = *(const v2f*)(sB + (wn * 32 + j * 16 + r) * LDSROW + kk + 2 * hi);
            #pragma unroll
            for (int i = 0; i < 4; i++)
                #pragma unroll
                for (int j = 0; j < 2; j++)
                    acc[i][j] = __builtin_amdgcn_wmma_f32_16x16x4_f32(
                        false, a[i], false, b[j], (short)0, acc[i][j], false, false);
        }

        async_wait_all();      // next chunk resident
        __syncthreads();       // everyone done reading current buffer
    }

    // ---- store + bias.  C/D layout: VGPR v -> row (v + 8*hi), col = lane&15
    #pragma unroll
    for (int j = 0; j < 2; j++) {
        int col = nBase + wn * 32 + j * 16 + r;
        float bv = bias[col];
        #pragma unroll
        for (int i = 0; i < 4; i++) {
            #pragma unroll
            for (int v = 0; v < 8; v++) {
                int row = mBase + wm * 64 + i * 16 + hi * 8 + v;
                if (row < M) C[(size_t)row * Nout + col] = acc[i][j][v] + bv;
            }
        }
    }
}

// ---------------------------------------------------------------------------
// helpers
// ---------------------------------------------------------------------------
__global__ void fill_zero(float* __restrict__ p, size_t n) {
    size_t i      = (size_t)blockIdx.x * blockDim.x + threadIdx.x;
    size_t stride = (size_t)gridDim.x * blockDim.x;
    for (; i < n; i += stride) p[i] = 0.0f;
}

__global__ void degree_accum(const int* __restrict__ dst, float* __restrict__ deg, int E) {
    int i = blockIdx.x * blockDim.x + threadIdx.x;
    if (i < E) atomicAdd(&deg[dst[i]], 1.0f);
}

__global__ void invert_deg(float* __restrict__ deg, int n) {
    int i = blockIdx.x * blockDim.x + threadIdx.x;
    if (i < n) deg[i] = 1.0f / fmaxf(deg[i], 1.0f);
}

// one wave32 per edge: agg[dst] += proj[src] * invdeg[dst]   (mean, L2 atomics)
__global__ __launch_bounds__(256)
void scatter_mean(const float* __restrict__ proj, const int* __restrict__ src,
                  const int* __restrict__ dst, const float* __restrict__ invdeg,
                  float* __restrict__ agg, int E)
{
    int wave = (int)((blockIdx.x * blockDim.x + threadIdx.x) >> 5);
    int lane = threadIdx.x & 31;
    if (wave >= E) return;
    int s = src[wave];
    int d = dst[wave];
    float w = invdeg[d];
    const v4f* ps = (const v4f*)(proj + (size_t)s * D);
    float*     pd = agg + (size_t)d * D;
    #pragma unroll
    for (int j = 0; j < 2; j++) {
        v4f v = ps[lane + 32 * j];          // b128 load, L2-resident
        int c = (lane + 32 * j) * 4;
        atomicAdd(&pd[c + 0], v[0] * w);
        atomicAdd(&pd[c + 1], v[1] * w);
        atomicAdd(&pd[c + 2], v[2] * w);
        atomicAdd(&pd[c + 3], v[3] * w);
    }
}

// fused GRUCell + LayerNorm: one wave32 per node, shuffle-only reduction
__global__ __launch_bounds__(256)
void gru_layernorm(const float* __restrict__ gi, const float* __restrict__ gh,
                   const float* __restrict__ hin, const float* __restrict__ gamma,
                   const float* __restrict__ beta, float* __restrict__ hout, int Mn)
{
    int row  = (int)((blockIdx.x * blockDim.x + threadIdx.x) >> 5);
    int lane = threadIdx.x & 31;
    if (row >= Mn) return;

    const float* gir = gi  + (size_t)row * 3 * D;
    const float* ghr = gh  + (size_t)row * 3 * D;
    const float* hr  = hin + (size_t)row * D;

    float h[8];
    float s1 = 0.0f, s2 = 0.0f;
    #pragma unroll
    for (int j = 0; j < 8; j++) {
        int c = lane + 32 * j;
        float i_r = gir[c], i_z = gir[D + c], i_n = gir[2 * D + c];
        float h_r = ghr[c], h_z = ghr[D + c], h_n = ghr[2 * D + c];
        float rg = 1.0f / (1.0f + __expf(-(i_r + h_r)));
        float zg = 1.0f / (1.0f + __expf(-(i_z + h_z)));
        float ng = tanhf(i_n + rg * h_n);
        h[j] = (1.0f - zg) * ng + zg * hr[c];
        s1 += h[j];
        s2 += h[j] * h[j];
    }
    #pragma unroll
    for (int off = 16; off > 0; off >>= 1) {
        s1 += __shfl_xor(s1, off, 32);
        s2 += __shfl_xor(s2, off, 32);
    }
    float mu   = s1 * (1.0f / D);
    float rstd = rsqrtf(s2 * (1.0f / D) - mu * mu + EPSLN);

    float* ho = hout + (size_t)row * D;
    #pragma unroll
    for (int j = 0; j < 8; j++) {
        int c = lane + 32 * j;
        ho[c] = (h[j] - mu) * rstd * gamma[c] + beta[c];
    }
}

// ---------------------------------------------------------------------------
// launch
// ---------------------------------------------------------------------------
extern "C" void kernel_launch(void* const* d_in, const int* in_sizes, int n_in,
                              void* d_out, int out_size, void* d_ws, size_t ws_size,
                              hipStream_t stream)
{
    const float* x = (const float*)d_in[0];
    const int*   srcs[4]; const int* dsts[4];
    const float* Ws[4];   const float* bs[4];
    for (int e = 0; e < 4; e++) {
        srcs[e] = (const int*)  d_in[1 + 4 * e];
        dsts[e] = (const int*)  d_in[2 + 4 * e];
        Ws[e]   = (const float*)d_in[3 + 4 * e];
        bs[e]   = (const float*)d_in[4 + 4 * e];
    }
    const float* W_ih  = (const float*)d_in[17];
    const float* W_hh  = (const float*)d_in[18];
    const float* b_ih  = (const float*)d_in[19];
    const float* b_hh  = (const float*)d_in[20];
    const float* gamma = (const float*)d_in[21];
    const float* beta  = (const float*)d_in[22];

    const int Mn = in_sizes[0] / D;   // nodes (50000)
    const int E  = in_sizes[1];       // edges per etype (300000)

    // workspace carve-up
    char*  ws  = (char*)d_ws;
    size_t off = 0;
    auto carve = [&](size_t bytes) -> float* {
        float* p = (float*)(ws + off);
        off += (bytes + 255) & ~(size_t)255;
        return p;
    };
    float* hA    = carve((size_t)Mn * D * 4);       //  51.2 MB
    float* proj  = carve((size_t)Mn * D * 4);       //  51.2 MB
    float* agg   = carve((size_t)Mn * D * 4);       //  51.2 MB
    float* gi    = carve((size_t)Mn * 3 * D * 4);   // 153.6 MB
    float* gh    = carve((size_t)Mn * 3 * D * 4);   // 153.6 MB
    float* invdg = carve((size_t)4 * Mn * 4);       //   0.8 MB
    (void)ws_size; (void)n_in; (void)out_size;

    // degrees are step-invariant: compute 1/max(deg,1) once for all 4 etypes
    {
        size_t n4 = (size_t)4 * Mn;
        fill_zero<<<1024, 256, 0, stream>>>(invdg, n4);
        for (int e = 0; e < 4; e++)
            degree_accum<<<(E + 255) / 256, 256, 0, stream>>>(dsts[e], invdg + (size_t)e * Mn, E);
        invert_deg<<<((int)n4 + 255) / 256, 256, 0, stream>>>(invdg, (int)n4);
    }

    const float* hin = x;
    for (int step = 0; step < STEPS; step++) {
        fill_zero<<<2048, 256, 0, stream>>>(agg, (size_t)Mn * D);

        for (int e = 0; e < 4; e++) {
            dim3 g((D + BN - 1) / BN, (Mn + BM - 1) / BM);
            gemm_bias_wmma<<<g, 256, 0, stream>>>(hin, Ws[e], bs[e], proj, Mn, D);
            scatter_mean<<<(E + 7) / 8, 256, 0, stream>>>(proj, srcs[e], dsts[e],
                                                          invdg + (size_t)e * Mn, agg, E);
        }
        {
            dim3 g((3 * D + BN - 1) / BN, (Mn + BM - 1) / BM);
            gemm_bias_wmma<<<g, 256, 0, stream>>>(agg, W_ih, b_ih, gi, Mn, 3 * D);
            gemm_bias_wmma<<<g, 256, 0, stream>>>(hin, W_hh, b_hh, gh, Mn, 3 * D);
        }

        float* hout = (step == STEPS - 1) ? (float*)d_out : hA;
        gru_layernorm<<<(Mn * 32 + 255) / 256, 256, 0, stream>>>(gi, gh, hin, gamma, beta, hout, Mn);
        hin = hA;   // steps 1,2 run in-place on hA (all reads precede the write)
    }
}